// Polarization_34531537060559
// MI455X (gfx1250) — compile-verified
//
#include <hip/hip_runtime.h>
#include <hip/hip_bf16.h>
#include <math.h>

#define TPB 256
#define DAMP 0.39f

typedef __attribute__((ext_vector_type(2))) float v2f;
typedef __attribute__((ext_vector_type(8))) float v8f;

// ---------------------------------------------------------------------------
// Wave32 sum using V_WMMA_F32_16X16X4_F32 against an all-ones B matrix.
// Each lane stores v/2 in both A regs -> D row m = p[m] + p[m+16] (all cols).
// Sum of own 8 D regs + shfl_xor(16) partner = full 32-lane sum in every lane.
// ---------------------------------------------------------------------------
__device__ __forceinline__ float wave_sum_wmma(float v) {
  v2f a; a[0] = v * 0.5f; a[1] = v * 0.5f;
  v2f b; b[0] = 1.0f;     b[1] = 1.0f;
  v8f c = {0.f, 0.f, 0.f, 0.f, 0.f, 0.f, 0.f, 0.f};
  c = __builtin_amdgcn_wmma_f32_16x16x4_f32(false, a, false, b,
                                            (short)0, c, false, false);
  float s = ((c[0] + c[1]) + (c[2] + c[3])) + ((c[4] + c[5]) + (c[6] + c[7]));
  s += __shfl_xor(s, 16, 32);
  return s;
}

__device__ __forceinline__ float block_sum(float v) {
  __shared__ float sm[TPB / 32];
  const int lane = threadIdx.x & 31;
  const int wid  = threadIdx.x >> 5;
  float w = wave_sum_wmma(v);
  if (lane == 0) sm[wid] = w;
  __syncthreads();
  float r = 0.0f;
  if (wid == 0) {
    float t = (lane < (TPB / 32)) ? sm[lane] : 0.0f;
    r = wave_sum_wmma(t);
  }
  return r;  // valid in wave 0
}

// ---------------------------------------------------------------------------
// Utility kernels (all CG state vectors are N x float4, pad lane == 0)
// ---------------------------------------------------------------------------
__global__ void k_zero(float* __restrict__ p, int n) {
  int i = blockIdx.x * blockDim.x + threadIdx.x;
  if (i < n) p[i] = 0.0f;
}

__global__ void k_ainv(const float* __restrict__ alpha, float* __restrict__ ainv, int n) {
  int i = blockIdx.x * blockDim.x + threadIdx.x;
  if (i < n) ainv[i] = 1.0f / fmaxf(alpha[i], 1e-12f);
}

__global__ void k_copy2(const float* __restrict__ src, float* __restrict__ r,
                        float* __restrict__ p, int n) {
  int i = blockIdx.x * blockDim.x + threadIdx.x;
  if (i < n) { float v = src[i]; r[i] = v; p[i] = v; }
}

// dot product with WMMA wave reduction; accumulates into *out (pre-zeroed).
// Pad lanes hold exact zeros, so N4-length dots equal the N3 reference dots.
__global__ void k_dot(const float* __restrict__ a, const float* __restrict__ b,
                      int n, float* __restrict__ out) {
  int i = blockIdx.x * blockDim.x + threadIdx.x;
  float v = (i < n) ? a[i] * b[i] : 0.0f;
  float s = block_sum(v);
  if (threadIdx.x == 0) atomicAdd(out, s);
}

// ---------------------------------------------------------------------------
// One-time edge precompute: E_perm scatter + per-edge {r_hat, damped/r^3}
// ---------------------------------------------------------------------------
__global__ void k_edge_setup(const float* __restrict__ pos,
                             const float* __restrict__ charge,
                             const float* __restrict__ alpha,
                             const int2* __restrict__ ei,
                             float4* __restrict__ ef,
                             float* __restrict__ Eperm4, int E) {
  int e = blockIdx.x * blockDim.x + threadIdx.x;
  if (e >= E) return;
  __builtin_prefetch(&ei[e + 8192], 0, 1);  // global_prefetch_b8 ahead of stream
  int2 sd = ei[e];
  const int s = sd.x, d = sd.y;
  float rx = pos[3 * d + 0] - pos[3 * s + 0];
  float ry = pos[3 * d + 1] - pos[3 * s + 1];
  float rz = pos[3 * d + 2] - pos[3 * s + 2];
  float rn = fmaxf(sqrtf(rx * rx + ry * ry + rz * rz), 1e-8f);
  float inv = 1.0f / rn;
  float hx = rx * inv, hy = ry * inv, hz = rz * inv;
  // field_contrib = charge[dst] * r_hat / r_norm
  float q = charge[d] * inv;
  atomicAdd(&Eperm4[4 * s + 0], q * hx);
  atomicAdd(&Eperm4[4 * s + 1], q * hy);
  atomicAdd(&Eperm4[4 * s + 2], q * hz);
  // Thole damping
  float r3 = rn * rn * rn;
  float as = fmaxf(alpha[s], 1e-12f);
  float ad = fmaxf(alpha[d], 1e-12f);
  float u = DAMP * rn / fmaxf(powf(as * ad, 1.0f / 6.0f), 1e-12f);
  float damping = 1.0f - (1.0f + u + 0.5f * u * u) * expf(-u);
  float dir3 = damping / fmaxf(r3, 1e-12f);
  ef[e] = make_float4(hx, hy, hz, dir3);
}

// ---------------------------------------------------------------------------
// CG matvec:  Av = alpha_inv .* v  -  segment_sum(dir3*(3*dot*r_hat - v_dst))
// ---------------------------------------------------------------------------
__global__ void k_diag(const float* __restrict__ ainv, const float* __restrict__ p,
                       float* __restrict__ Ap, int n4) {
  int i = blockIdx.x * blockDim.x + threadIdx.x;
  if (i < n4) Ap[i] = ainv[i >> 2] * p[i];   // pad lane: ainv * 0 = 0
}

__global__ void k_edge_matvec(const float4* __restrict__ ef,
                              const int2* __restrict__ ei,
                              const float4* __restrict__ v4,
                              float* __restrict__ Av4, int E) {
  int e = blockIdx.x * blockDim.x + threadIdx.x;
  if (e >= E) return;
  __builtin_prefetch(&ef[e + 4096], 0, 1);  // stream-ahead prefetch (speculative)
  __builtin_prefetch(&ei[e + 8192], 0, 1);
  int2 sd = ei[e];
  float4 f = ef[e];
  float4 vd = v4[sd.y];                     // single aligned global_load_b128 gather
  const float c3 = 3.0f * (f.x * vd.x + f.y * vd.y + f.z * vd.z);
  // Av[src] -= t_contrib
  atomicAdd(&Av4[4 * sd.x + 0], -f.w * (c3 * f.x - vd.x));
  atomicAdd(&Av4[4 * sd.x + 1], -f.w * (c3 * f.y - vd.y));
  atomicAdd(&Av4[4 * sd.x + 2], -f.w * (c3 * f.z - vd.z));
}

// ---------------------------------------------------------------------------
// CG scalar control (mirrors reference halt/conv semantics)
// sc: 0=rs_old 1=bnorm2 2=pAp 3=rs_new 4=a 5=beta 6=done 7=halt 8=stop
// ---------------------------------------------------------------------------
__global__ void k_scalar_init(float* sc) {
  float rs = sc[0];
  sc[1] = rs;
  sc[6] = (sqrtf(rs) < 1e-12f) ? 1.0f : 0.0f;
  sc[2] = 0.0f; sc[3] = 0.0f; sc[4] = 0.0f; sc[5] = 0.0f;
  sc[7] = 0.0f; sc[8] = 0.0f;
}

__global__ void k_scalar_a(float* sc) {
  float pAp  = sc[2];
  float halt = (sc[6] != 0.0f || fabsf(pAp) < 1e-12f) ? 1.0f : 0.0f;
  float den  = (fabsf(pAp) < 1e-12f) ? 1.0f : pAp;
  sc[4] = sc[0] / den;  // a = rs_old / den
  sc[7] = halt;
  sc[3] = 0.0f;         // zero rs_new accumulator
}

__global__ void k_scalar_b(float* sc) {
  float rs_new = sc[3];
  float bnorm  = sqrtf(sc[1]);
  float conv   = (sqrtf(rs_new) / fmaxf(bnorm, 1e-12f) < 1e-6f) ? 1.0f : 0.0f;
  float stop   = (sc[7] != 0.0f || conv != 0.0f) ? 1.0f : 0.0f;
  float rs_old = sc[0];
  sc[5] = rs_new / ((rs_old == 0.0f) ? 1.0f : rs_old);  // beta
  sc[8] = stop;
  sc[6] = stop;                                         // done = stop
  if (stop == 0.0f) sc[0] = rs_new;
  sc[2] = 0.0f;  // zero pAp accumulator for next iteration
}

__global__ void k_axpy(float* __restrict__ x, float* __restrict__ r,
                       const float* __restrict__ p, const float* __restrict__ Ap,
                       const float* __restrict__ sc, int n4) {
  int i = blockIdx.x * blockDim.x + threadIdx.x;
  if (i >= n4) return;
  if (sc[7] == 0.0f) {
    float a = sc[4];
    x[i] += a * p[i];
    r[i] -= a * Ap[i];
  }
}

__global__ void k_pupd(float* __restrict__ p, const float* __restrict__ r,
                       const float* __restrict__ sc, int n4) {
  int i = blockIdx.x * blockDim.x + threadIdx.x;
  if (i >= n4) return;
  if (sc[8] == 0.0f) p[i] = r[i] + sc[5] * p[i];
}

// ---------------------------------------------------------------------------
// Final per-atom energy -> per-graph bins (two b128 loads per atom)
// ---------------------------------------------------------------------------
__global__ void k_energy(const float4* __restrict__ mu4, const float4* __restrict__ Ep4,
                         const int* __restrict__ batch, float* __restrict__ out, int n) {
  int i = blockIdx.x * blockDim.x + threadIdx.x;
  if (i >= n) return;
  float4 m = mu4[i];
  float4 b = Ep4[i];
  float e = -0.5f * (m.x * b.x + m.y * b.y + m.z * b.z);
  atomicAdd(&out[batch[i]], e);
}

// ---------------------------------------------------------------------------
extern "C" void kernel_launch(void* const* d_in, const int* in_sizes, int n_in,
                              void* d_out, int out_size, void* d_ws, size_t ws_size,
                              hipStream_t stream) {
  const float* pos    = (const float*)d_in[0];
  const float* charge = (const float*)d_in[1];
  const float* alpha  = (const float*)d_in[2];
  const int*   batch  = (const int*)d_in[3];
  const int2*  ei     = (const int2*)d_in[4];   // [E][2] = {src, dst}
  float* out = (float*)d_out;

  const int N  = in_sizes[1];
  const int E  = in_sizes[4] / 2;
  const int N4 = 4 * N;   // padded float4 layout, pad lane kept at 0

  // workspace layout (all 16B aligned)
  char* ws = (char*)d_ws;
  float4* ef   = (float4*)ws;  ws += (size_t)E * 16;
  float* Eperm = (float*)ws;   ws += (size_t)N4 * 4;
  float* x     = (float*)ws;   ws += (size_t)N4 * 4;
  float* r     = (float*)ws;   ws += (size_t)N4 * 4;
  float* p     = (float*)ws;   ws += (size_t)N4 * 4;
  float* Ap    = (float*)ws;   ws += (size_t)N4 * 4;
  float* ainv  = (float*)ws;   ws += (((size_t)N * 4 + 255) / 256) * 256;
  float* sc    = (float*)ws;   // 16 control scalars

  const dim3 blk(TPB);
  const int gN4 = (N4 + TPB - 1) / TPB;
  const int gN  = (N  + TPB - 1) / TPB;
  const int gE  = (E  + TPB - 1) / TPB;

  // setup
  k_zero<<<gN4, blk, 0, stream>>>(Eperm, N4);
  k_zero<<<gN4, blk, 0, stream>>>(x, N4);
  k_zero<<<1, 64, 0, stream>>>(sc, 16);
  k_ainv<<<gN, blk, 0, stream>>>(alpha, ainv, N);
  k_edge_setup<<<gE, blk, 0, stream>>>(pos, charge, alpha, ei, ef, Eperm, E);
  k_dot<<<gN4, blk, 0, stream>>>(Eperm, Eperm, N4, &sc[0]);   // rs_old = b.b
  k_copy2<<<gN4, blk, 0, stream>>>(Eperm, r, p, N4);
  k_scalar_init<<<1, 1, 0, stream>>>(sc);

  // 50 fixed CG iterations (halt/conv handled via device scalars)
  for (int it = 0; it < 50; ++it) {
    k_diag<<<gN4, blk, 0, stream>>>(ainv, p, Ap, N4);
    k_edge_matvec<<<gE, blk, 0, stream>>>(ef, ei, (const float4*)p, Ap, E);
    k_dot<<<gN4, blk, 0, stream>>>(p, Ap, N4, &sc[2]);        // pAp
    k_scalar_a<<<1, 1, 0, stream>>>(sc);
    k_axpy<<<gN4, blk, 0, stream>>>(x, r, p, Ap, sc, N4);
    k_dot<<<gN4, blk, 0, stream>>>(r, r, N4, &sc[3]);         // rs_new
    k_scalar_b<<<1, 1, 0, stream>>>(sc);
    k_pupd<<<gN4, blk, 0, stream>>>(p, r, sc, N4);
  }

  // energy reduction into 256 graph bins
  k_zero<<<(out_size + TPB - 1) / TPB, blk, 0, stream>>>(out, out_size);
  k_energy<<<gN, blk, 0, stream>>>((const float4*)x, (const float4*)Eperm, batch, out, N);
}